// MAMBA2D_15616501088664
// MI455X (gfx1250) — compile-verified
//
#include <hip/hip_runtime.h>
#include <hip/hip_bf16.h>

typedef float v2f __attribute__((ext_vector_type(2)));
typedef float v8f __attribute__((ext_vector_type(8)));

// ---- problem constants ----
#define CC    192
#define DII   384
#define DSS   16
#define DTR   12
#define LSEQ  64
#define NSEQ  256
#define MROWS 16384          // NSEQ * LSEQ per direction
#define TWODI 768

// pixel index (into flat B*H*W) for direction d, flat sequence-row m = n*64 + l.
// d0: horiz fwd, d1: horiz bwd, d2: vert fwd, d3: vert bwd. Each map is a
// permutation, so it serves both the input gather and the output scatter
// (flipped dirs are automatically un-flipped on scatter).
__device__ __forceinline__ int pixmap(int d, int m) {
  int n = m >> 6, l = m & 63;
  if (d == 0) return (n << 6) + l;
  if (d == 1) return (n << 6) + (63 - l);
  if (d == 2) return ((n >> 6) << 12) + (l << 6) + (n & 63);
  return ((n >> 6) << 12) + ((63 - l) << 6) + (n & 63);
}

// ---------------------------------------------------------------------------
// Generic fp32 WMMA GEMM: D[M,N] = A[M,K] * W[N,K]^T (+bias)
// Block tile 128x64, 8 waves, each wave: 16 rows x 64 cols (4 v_wmma tiles).
// W chunks (KCx64) staged in double-buffered LDS via GLOBAL_LOAD_ASYNC_TO_LDS
// (ASYNCcnt), overlapping the next chunk's copy with the current chunk's
// wmma work. B fragments read back as (k,k+1) float2 pairs -> ds_load_b64.
// mapA: gather A rows through pixmap(d,row); mapD: scatter D rows through
// pixmap and offset columns into the 4C concat layout [d2|d3|d0|d1].
// ---------------------------------------------------------------------------
#define BM 128
#define BN 64
#define KC 96

__global__ __launch_bounds__(256)
void wmma_gemm_kernel(const float* __restrict__ A, const float* __restrict__ W,
                      float* __restrict__ D, const float* __restrict__ bias,
                      int M, int N, int K, int lda, int ldd,
                      long strideA, long strideW, long strideD,
                      int mapA, int mapD)
{
  __shared__ float2 bs[2][(KC / 2) * BN];       // 2 x 24 KB (double buffer)
  const int tid  = threadIdx.x;
  const int wave = tid >> 5;
  const int lane = tid & 31;
  const int half = lane >> 4;                   // K-pair selector
  const int lm   = lane & 15;                   // M (A) / N (B,D) within tile
  const int d     = blockIdx.z;
  const int mBase = blockIdx.x * BM;
  const int nBase = blockIdx.y * BN;

  const int rowA  = mBase + wave * 16 + lm;
  const long arow = mapA ? (long)pixmap(d, rowA) : (long)rowA;
  const float* Ap = A + strideA * d + arow * (long)lda;

  // Issue async copies of W chunk kc into LDS buffer `buf`.
  // bs[buf][kk*64 + j] = { B[k0+2kk][nBase+j], B[k0+2kk+1][nBase+j] }
  // Out-of-range columns are clamped (their accumulators are never stored).
  auto issue_chunk = [&](int kc, int buf) {
    const int k0 = kc * KC;
    for (int idx = tid; idx < (KC / 2) * BN; idx += 256) {
      int kk = idx >> 6;
      int j  = idx & 63;
      int gj = nBase + j;
      if (gj >= N) gj = N - 1;
      const float* wp = W + strideW * d + (long)gj * K + k0 + kk * 2;
      unsigned lds           = (unsigned)(size_t)&bs[buf][idx];
      unsigned long long ga  = (unsigned long long)(size_t)wp;
      asm volatile("global_load_async_to_lds_b64 %0, %1, off"
                   :: "v"(lds), "v"(ga) : "memory");
    }
  };

  v8f acc[4];
#pragma unroll
  for (int i = 0; i < 4; ++i) acc[i] = 0.f;

  const int nChunks = K / KC;
  issue_chunk(0, 0);
  for (int kc = 0; kc < nChunks; ++kc) {
    // Wait for this wave's async copies of chunk kc, then barrier so every
    // wave's copies are visible in LDS.
    asm volatile("s_wait_asynccnt 0x0" ::: "memory");
    __syncthreads();
    // Kick off the next chunk's copies into the other buffer (overlaps wmma).
    if (kc + 1 < nChunks) issue_chunk(kc + 1, (kc + 1) & 1);

    const int k0 = kc * KC;
    const float2* bb = bs[kc & 1];
    if (kc + 1 < nChunks)
      __builtin_prefetch(Ap + k0 + KC, 0, 0);   // global_prefetch next A chunk
#pragma unroll
    for (int t = 0; t < KC / 4; ++t) {
      const int kl = t * 4;
      // A fragment (16x4): lane = M, half picks K pair -> one b64 global load
      v2f a = *(const v2f*)(Ap + k0 + kl + 2 * half);
      const int kk = (kl >> 1) + half;
      const float2* brow = &bb[kk * BN + lm];
      v2f b0 = *(const v2f*)&brow[0];
      v2f b1 = *(const v2f*)&brow[16];
      v2f b2 = *(const v2f*)&brow[32];
      v2f b3 = *(const v2f*)&brow[48];
      acc[0] = __builtin_amdgcn_wmma_f32_16x16x4_f32(false, a, false, b0, (short)0, acc[0], false, false);
      acc[1] = __builtin_amdgcn_wmma_f32_16x16x4_f32(false, a, false, b1, (short)0, acc[1], false, false);
      acc[2] = __builtin_amdgcn_wmma_f32_16x16x4_f32(false, a, false, b2, (short)0, acc[2], false, false);
      acc[3] = __builtin_amdgcn_wmma_f32_16x16x4_f32(false, a, false, b3, (short)0, acc[3], false, false);
    }
  }

  // D fragment: vgpr v holds row M = v + 8*half, lane%16 = col
#pragma unroll
  for (int nt = 0; nt < 4; ++nt) {
#pragma unroll
    for (int v = 0; v < 8; ++v) {
      const int row = mBase + wave * 16 + v + 8 * half;
      const int col = nBase + nt * 16 + lm;
      if (col < N) {
        float val = acc[nt][v];
        if (bias) val += bias[col];
        if (mapD) {
          const int coff = ((d + 2) & 3) * 192;   // d2->0, d3->192, d0->384, d1->576
          D[(long)pixmap(d, row) * ldd + coff + col] = val;
        } else {
          D[strideD * d + (long)row * ldd + col] = val;
        }
      }
    }
  }
}

// ---------------------------------------------------------------------------
// Causal depthwise conv1d(4) + bias + SiLU:  XZ[:, :384] -> UC
// ---------------------------------------------------------------------------
__global__ __launch_bounds__(256)
void conv_silu_kernel(const float* __restrict__ XZ, const float* __restrict__ cw,
                      const float* __restrict__ cb, float* __restrict__ UC)
{
  const long idx = (long)blockIdx.x * 256 + threadIdx.x;
  const long total = 4L * MROWS * DII;
  if (idx >= total) return;
  const int  ch = (int)(idx % DII);
  const long md = idx / DII;                 // d*16384 + m
  const int  m  = (int)(md % MROWS);
  const int  l  = m & 63;
  const long rowBase = md - l;               // start of this sequence in row space
  const int  dch = (int)(md / MROWS) * DII + ch;

  float accv = cb[dch];
  const float* cwp = cw + (long)dch * 4;
#pragma unroll
  for (int t = 0; t < 4; ++t) {
    const int ls = l - 3 + t;
    float u = 0.f;
    if (ls >= 0) u = XZ[(rowBase + ls) * TWODI + ch];
    accv += cwp[t] * u;
  }
  UC[idx] = accv / (1.f + __expf(-accv));    // SiLU
}

// ---------------------------------------------------------------------------
// Fused dt_proj + softplus + selective scan + D-skip + SiLU(z) gate.
// One block per (dir, sequence); one thread per inner channel; 16-state h in
// VGPRs; per-step x_dbl row (dt|B|C = 44 floats) staged in LDS.
// Writes gated y in place over UC.
// ---------------------------------------------------------------------------
__global__ __launch_bounds__(384)
void scan_kernel(const float* __restrict__ XZ, float* __restrict__ UC,
                 const float* __restrict__ XDBL,
                 const float* __restrict__ dtw, const float* __restrict__ dtb,
                 const float* __restrict__ Alog, const float* __restrict__ Dv)
{
  const int n  = blockIdx.x;                 // sequence 0..255
  const int d  = blockIdx.y;                 // direction 0..3
  const int ch = threadIdx.x;                // 0..383
  const int dch = d * DII + ch;
  __shared__ float sh[48];                   // [0,12) dt  [12,28) B  [28,44) C

  float wdt[DTR];
#pragma unroll
  for (int r = 0; r < DTR; ++r) wdt[r] = dtw[(long)dch * DTR + r];
  float Ar[DSS];
#pragma unroll
  for (int s = 0; s < DSS; ++s) Ar[s] = -__expf(Alog[(long)dch * DSS + s]);
  const float bdt = dtb[dch];
  const float dsk = Dv[dch];
  float h[DSS];
#pragma unroll
  for (int s = 0; s < DSS; ++s) h[s] = 0.f;

  const long mBase = (long)d * MROWS + (long)n * LSEQ;
  for (int l = 0; l < LSEQ; ++l) {
    if (ch < 44) sh[ch] = XDBL[(mBase + l) * 44 + ch];
    __syncthreads();
    const float u = UC[(mBase + l) * DII + ch];
    const float z = XZ[(mBase + l) * TWODI + DII + ch];
    float dtv = bdt;
#pragma unroll
    for (int r = 0; r < DTR; ++r) dtv += wdt[r] * sh[r];
    const float delta = (dtv > 20.f) ? dtv : __logf(1.f + __expf(dtv));
    const float du = delta * u;
    float y = 0.f;
#pragma unroll
    for (int s = 0; s < DSS; ++s) {
      const float dA = __expf(delta * Ar[s]);
      h[s] = dA * h[s] + du * sh[12 + s];
      y += h[s] * sh[28 + s];
    }
    const float sz = z / (1.f + __expf(-z)); // SiLU(z)
    UC[(mBase + l) * DII + ch] = (y + dsk * u) * sz;
    __syncthreads();
  }
}

// ---------------------------------------------------------------------------
extern "C" void kernel_launch(void* const* d_in, const int* in_sizes, int n_in,
                              void* d_out, int out_size, void* d_ws, size_t ws_size,
                              hipStream_t stream)
{
  const float* x    = (const float*)d_in[0];
  const float* inw  = (const float*)d_in[1];   // [4, 768, 192]
  const float* cw   = (const float*)d_in[2];   // [4, 384, 4]
  const float* cb   = (const float*)d_in[3];   // [4, 384]
  const float* xw   = (const float*)d_in[4];   // [4, 44, 384]
  const float* dtw  = (const float*)d_in[5];   // [4, 384, 12]
  const float* dtb  = (const float*)d_in[6];   // [4, 384]
  const float* Alog = (const float*)d_in[7];   // [4, 384, 16]
  const float* Dv   = (const float*)d_in[8];   // [4, 384]
  const float* ow   = (const float*)d_in[9];   // [4, 192, 384]
  const float* fcw  = (const float*)d_in[10];  // [192, 768]
  const float* fcb  = (const float*)d_in[11];  // [192]
  float* out = (float*)d_out;                  // [16384, 192]

  char* ws = (char*)d_ws;
  float* XZ   = (float*)(ws);                  // [4,16384,768] : 201326592 B
  float* UC   = (float*)(ws + 201326592L);     // [4,16384,384] : 100663296 B
  float* XDBL = (float*)(ws + 301989888L);     // [4,16384,44]  :  11534336 B
  float* OUT4 = (float*)(ws + 313524224L);     // [16384,768]   :  50331648 B

  const dim3 blk(256);

  // 1) in_proj: XZ = gather(x) @ in_w^T        (K=192, N=768, per dir)
  wmma_gemm_kernel<<<dim3(128, 12, 4), blk, 0, stream>>>(
      x, inw, XZ, nullptr, MROWS, TWODI, CC, CC, TWODI,
      0L, (long)TWODI * CC, (long)MROWS * TWODI, 1, 0);

  // 2) depthwise causal conv + SiLU: UC = silu(conv(u) + b)
  {
    const long total = 4L * MROWS * DII;
    conv_silu_kernel<<<dim3((unsigned)((total + 255) / 256)), blk, 0, stream>>>(
        XZ, cw, cb, UC);
  }

  // 3) x_proj: XDBL = UC @ x_w^T               (K=384, N=44, per dir)
  wmma_gemm_kernel<<<dim3(128, 1, 4), blk, 0, stream>>>(
      UC, xw, XDBL, nullptr, MROWS, DTR + 2 * DSS, DII, DII, DTR + 2 * DSS,
      (long)MROWS * DII, (long)(DTR + 2 * DSS) * DII, (long)MROWS * (DTR + 2 * DSS), 0, 0);

  // 4) fused dt_proj + selective scan + gate (in place over UC)
  scan_kernel<<<dim3(NSEQ, 4), dim3(384), 0, stream>>>(
      XZ, UC, XDBL, dtw, dtb, Alog, Dv);

  // 5) out_proj, scattered into the [d2|d3|d0|d1] 4C concat (K=384, N=192)
  wmma_gemm_kernel<<<dim3(128, 3, 4), blk, 0, stream>>>(
      UC, ow, OUT4, nullptr, MROWS, CC, DII, DII, 4 * CC,
      (long)MROWS * DII, (long)CC * DII, 0L, 0, 1);

  // 6) fc: out = OUT4 @ fc_w^T + fc_b          (K=768, N=192)
  wmma_gemm_kernel<<<dim3(128, 3, 1), blk, 0, stream>>>(
      OUT4, fcw, out, fcb, MROWS, CC, 4 * CC, 4 * CC, CC,
      0L, 0L, 0L, 0, 0);
}